// LOA_BFGS_Model_70652212019446
// MI455X (gfx1250) — compile-verified
//
#include <hip/hip_runtime.h>

#define PDIM 256

typedef __attribute__((ext_vector_type(2))) float v2f;
typedef __attribute__((ext_vector_type(4))) float v4f;
typedef __attribute__((ext_vector_type(8))) float v8f;

#if defined(__gfx1250__) && __has_builtin(__builtin_amdgcn_wmma_f32_16x16x4_f32)
#define USE_WMMA 1
#else
#define USE_WMMA 0
#endif

// Per-wave 16-element MLP layer:  hout[o] = (relu?) sum_k W[o][k]*hin[k]
// "colvec" convention: register hin[k] holds the value for element/column
// c = (lane & 15) in EVERY lane (replicated across both lane halves).
//
// WMMA f32 16x16x4 usage: each step feeds K-slots {0,3} only (slots 1,2 = 0).
//   A (16x4, explicit ISA layout): A[m][0] -> lanes0-15 vgpr0 ; A[m][3] -> lanes16-31 vgpr1
//   B (4x16): K=0 -> vgpr0 lanes0-15 and K=3 -> vgpr1 lanes16-31 under both
//             consistent readings of the ISA B layout (ambiguous slots 1,2
//             are kept zero), so the pairing is layout-ambiguity proof.
//   D (16x16, explicit ISA layout): row m -> vgpr (m&7), lane group (m>>3).
//
// Weight reads are index-CLAMPED so the ds_load is unconditional (no
// s_and_saveexec around WMMA); masking to zero is a v_cndmask select.
template <int IN, int OUT, bool RELU>
__device__ __forceinline__ void mlp_layer(const float* __restrict__ W,
                                          const float (&hin)[IN],
                                          float (&hout)[OUT],
                                          const int lane) {
#if USE_WMMA
  const bool lo = lane < 16;
  const int m = lane & 15;
  const int mc = (m < OUT) ? m : (OUT - 1);            // clamped, always valid
  v8f c0 = {0.f, 0.f, 0.f, 0.f, 0.f, 0.f, 0.f, 0.f};
  v8f c1 = {0.f, 0.f, 0.f, 0.f, 0.f, 0.f, 0.f, 0.f};
#pragma unroll
  for (int k = 0; k < IN; k += 2) {
    const bool haveB = (k + 1) < IN;
    const int kb = haveB ? (k + 1) : k;                // clamped
    v2f b;
    b.x = lo ? hin[k] : 0.0f;                          // K=0 row (lanes 0-15)
    b.y = (haveB && !lo) ? hin[kb] : 0.0f;             // K=3 row (lanes 16-31)
    {
      const float wa = W[mc * IN + k];                 // unconditional ds_load
      const float wb = W[mc * IN + kb];
      v2f a;
      a.x = (lo && m < OUT) ? wa : 0.0f;               // A[m][0]
      a.y = (haveB && !lo && m < OUT) ? wb : 0.0f;     // A[m][3]
      c0 = __builtin_amdgcn_wmma_f32_16x16x4_f32(false, a, false, b,
                                                 (short)0, c0, false, false);
    }
    if (OUT > 16) {
      const int mm = m + 16;
      const int mmc = (mm < OUT) ? mm : (OUT - 1);
      const float wa = W[mmc * IN + k];
      const float wb = W[mmc * IN + kb];
      v2f a;
      a.x = (lo && mm < OUT) ? wa : 0.0f;
      a.y = (haveB && !lo && mm < OUT) ? wb : 0.0f;
      c1 = __builtin_amdgcn_wmma_f32_16x16x4_f32(false, a, false, b,
                                                 (short)0, c1, false, false);
    }
  }
  // D readout -> replicated colvec registers
#pragma unroll
  for (int o = 0; o < OUT; ++o) {
    const int om = (o < 16) ? o : (o - 16);
    float v = (o < 16) ? c0[om & 7] : c1[om & 7];
    const int src = ((om & 8) ? 16 : 0) + (lane & 15);
    float r = __shfl(v, src, 32);
    hout[o] = RELU ? fmaxf(r, 0.0f) : r;
  }
#else
  (void)lane;
#pragma unroll
  for (int o = 0; o < OUT; ++o) {
    float acc = 0.0f;
#pragma unroll
    for (int k = 0; k < IN; ++k) acc = fmaf(W[o * IN + k], hin[k], acc);
    hout[o] = RELU ? fmaxf(acc, 0.0f) : acc;
  }
#endif
}

// One block per trajectory n. 256 threads = 8 waves; each wave handles two
// 16-column tiles for the WMMA MLP phases.
__global__ __launch_bounds__(256) void loa_bfgs_fused(
    const float* __restrict__ grad, const float* __restrict__ gradm1,
    const float* __restrict__ dm1, const float* __restrict__ pmat,
    const float* __restrict__ Wfs, const float* __restrict__ Wo1,
    const float* __restrict__ Wo2, const float* __restrict__ Wo3,
    const float* __restrict__ Wl1, const float* __restrict__ Wl2,
    const float* __restrict__ Wl3, float* __restrict__ dout) {
  __shared__ float2 dgg_s[PDIM];       // (grad - gradm1, grad)
  __shared__ float qndg_s[PDIM];
  __shared__ float bg_s[PDIM];
  __shared__ float of3_s[3][PDIM];
  __shared__ float out_s[PDIM];
  __shared__ float red_s[4][PDIM];
  __shared__ float wbuf[516];          // Wo1|Wo2|Wo3|Wfs|Wl1|Wl2|Wl3

  const int n = blockIdx.x;
  const int tid = threadIdx.x;
  const int lane = tid & 31;
  const int wave = tid >> 5;

  // --- stage weights in LDS (offsets: 0,18,90,126,132,204,492; total 516) ---
  for (int i = tid; i < 516; i += 256) {
    float w;
    if (i < 18)        w = Wo1[i];
    else if (i < 90)   w = Wo2[i - 18];
    else if (i < 126)  w = Wo3[i - 90];
    else if (i < 132)  w = Wfs[i - 126];
    else if (i < 204)  w = Wl1[i - 132];
    else if (i < 492)  w = Wl2[i - 204];
    else               w = Wl3[i - 492];
    wbuf[i] = w;
  }

  const float gp = grad[n * PDIM + tid];
  const float gm = gradm1[n * PDIM + tid];
  dgg_s[tid] = make_float2(gp - gm, gp);
  __syncthreads();

  // --- phase 1: dual batched matvec, one streaming (non-temporal) pass over
  //     M[n] (256 KB per block, 128 MB total -> HBM roofline ~5.5us) ---
  {
    const v4f* __restrict__ Mrow =
        (const v4f*)(pmat + (size_t)n * PDIM * PDIM + (size_t)tid * PDIM);
    float accd = 0.0f, accg = 0.0f;
#pragma unroll 8
    for (int q4 = 0; q4 < PDIM / 4; ++q4) {
      const v4f mv = __builtin_nontemporal_load(Mrow + q4);  // th:NT stream
      const float2 w0 = dgg_s[q4 * 4 + 0];
      const float2 w1 = dgg_s[q4 * 4 + 1];
      const float2 w2 = dgg_s[q4 * 4 + 2];
      const float2 w3 = dgg_s[q4 * 4 + 3];
      accd = fmaf(mv.x, w0.x, accd); accg = fmaf(mv.x, w0.y, accg);
      accd = fmaf(mv.y, w1.x, accd); accg = fmaf(mv.y, w1.y, accg);
      accd = fmaf(mv.z, w2.x, accd); accg = fmaf(mv.z, w2.y, accg);
      accd = fmaf(mv.w, w3.x, accd); accg = fmaf(mv.w, w3.y, accg);
    }
    qndg_s[tid] = accd;      // QNDG = M (g - g_-1)
    bg_s[tid] = -accg;       // Bg   = -M g  (= M v, GAMMA == 1)
  }
  __syncthreads();

  // --- phase 2a: outer MLP (3->6->12->3) per element, WMMA tiles of 16 ---
  float xr[2][3];
#pragma unroll
  for (int j = 0; j < 2; ++j) {
    const int p = (wave * 2 + j) * 16 + (lane & 15);
    const float x0 = qndg_s[p];
    const float x1 = dm1[n * PDIM + p];
    const float x2 = bg_s[p];
    xr[j][0] = x0; xr[j][1] = x1; xr[j][2] = x2;
    float h0[3] = {x0, x1, x2};
    float h1[6];  mlp_layer<3, 6, true>(wbuf + 0, h0, h1, lane);
    float h2[12]; mlp_layer<6, 12, true>(wbuf + 18, h1, h2, lane);
    float h3[3];  mlp_layer<12, 3, false>(wbuf + 90, h2, h3, lane);
    if (lane < 16) {
      of3_s[0][p] = h3[0]; of3_s[1][p] = h3[1]; of3_s[2][p] = h3[2];
    }
  }
  __syncthreads();
  for (int s = 128; s > 0; s >>= 1) {
    if (tid < s) {
      of3_s[0][tid] += of3_s[0][tid + s];
      of3_s[1][tid] += of3_s[1][tid + s];
      of3_s[2][tid] += of3_s[2][tid + s];
    }
    __syncthreads();
  }
  const float of0 = of3_s[0][0] * (1.0f / PDIM);
  const float of1 = of3_s[1][0] * (1.0f / PDIM);
  const float of2 = of3_s[2][0] * (1.0f / PDIM);

  // --- phase 2b: fullskip (6->1) + inner MLP (6->12->24->1) ---
#pragma unroll
  for (int j = 0; j < 2; ++j) {
    const int p = (wave * 2 + j) * 16 + (lane & 15);
    float x6[6] = {xr[j][0], xr[j][1], xr[j][2], of0, of1, of2};
    float fs[1];  mlp_layer<6, 1, false>(wbuf + 126, x6, fs, lane);
    float g1[12]; mlp_layer<6, 12, true>(wbuf + 132, x6, g1, lane);
    float g2[24]; mlp_layer<12, 24, true>(wbuf + 204, g1, g2, lane);
    float g3[1];  mlp_layer<24, 1, false>(wbuf + 492, g2, g3, lane);
    if (lane < 16) out_s[p] = g3[0] + fs[0];
  }
  __syncthreads();

  // --- phase 3: BFGS scalars + final d (no PxP matrix materialized) ---
  const float dgk = dgg_s[tid].x;                       // grad - gradm1
  const float vq = -dgg_s[tid].y;                       // v = -grad
  const float sec = dm1[n * PDIM + tid] - qndg_s[tid];  // dm1 - QNDG
  const float outp = out_s[tid];
  red_s[0][tid] = outp * dgk;   // denom
  red_s[1][tid] = sec * dgk;    // secant . DGK
  red_s[2][tid] = outp * vq;    // out . v
  red_s[3][tid] = sec * vq;     // secant . v
  __syncthreads();
  for (int s = 128; s > 0; s >>= 1) {
    if (tid < s) {
      red_s[0][tid] += red_s[0][tid + s];
      red_s[1][tid] += red_s[1][tid + s];
      red_s[2][tid] += red_s[2][tid + s];
      red_s[3][tid] += red_s[3][tid + s];
    }
    __syncthreads();
  }
  const float denom = red_s[0][0];
  const float inv = 1.0f / denom;
  const float coef = red_s[1][0] * inv;
  const float outv = red_s[2][0];
  const float secv = red_s[3][0];

  // d = M v + (1/denom)(sec*(out.v) + out*(sec.v) - coef*out*(out.v))
  dout[n * PDIM + tid] =
      bg_s[tid] + inv * (sec * outv + outp * secv - coef * outp * outv);
}

extern "C" void kernel_launch(void* const* d_in, const int* in_sizes, int n_in,
                              void* d_out, int out_size, void* d_ws,
                              size_t ws_size, hipStream_t stream) {
  (void)n_in; (void)out_size; (void)d_ws; (void)ws_size;
  const float* grad   = (const float*)d_in[0];
  const float* gradm1 = (const float*)d_in[1];
  const float* dm1    = (const float*)d_in[2];
  const float* pmat   = (const float*)d_in[3];
  const float* Wfs    = (const float*)d_in[4];
  const float* Wo1    = (const float*)d_in[5];
  const float* Wo2    = (const float*)d_in[6];
  const float* Wo3    = (const float*)d_in[7];
  const float* Wl1    = (const float*)d_in[8];
  const float* Wl2    = (const float*)d_in[9];
  const float* Wl3    = (const float*)d_in[10];
  const int N = in_sizes[0] / PDIM;  // 512

  loa_bfgs_fused<<<dim3(N), dim3(256), 0, stream>>>(
      grad, gradm1, dm1, pmat, Wfs, Wo1, Wo2, Wo3, Wl1, Wl2, Wl3,
      (float*)d_out);
}